// EdgeComp_44418551775898
// MI455X (gfx1250) — compile-verified
//
#include <hip/hip_runtime.h>

#define BATCH 8
#define NPTS  4096
#define DIM   64
#define KNN   20
#define ROWS  32      // rows per wave-block (one per lane)
#define CTILE 16      // columns per WMMA tile
#define NTILES (NPTS / CTILE)
#define LDSP  17      // padded row stride (floats) -> conflict-free

typedef __attribute__((ext_vector_type(2))) float v2f;
typedef __attribute__((ext_vector_type(4))) float v4f;
typedef __attribute__((ext_vector_type(8))) float v8f;

// ---------------- kernel 0: squared norms ----------------
__global__ void EdgeSq_kernel(const float* __restrict__ x, float* __restrict__ sq) {
  int p = blockIdx.x * blockDim.x + threadIdx.x;
  if (p >= BATCH * NPTS) return;
  const v4f* xr = (const v4f*)(x + (size_t)p * DIM);
  float s = 0.f;
#pragma unroll
  for (int i = 0; i < DIM / 4; ++i) {
    v4f v = xr[i];
    s += v.x * v.x + v.y * v.y + v.z * v.z + v.w * v.w;
  }
  sq[p] = s;
}

// ---------------- kernel 1: WMMA distances + fused top-20 ----------------
// One wave (32 lanes) per block; block owns 32 rows of one batch.
// key(i,j) = sq[j] - 2*inner(i,j)   (same ranking as full sq distance)
__launch_bounds__(32)
__global__ void EdgeKnn_kernel(const float* __restrict__ x,
                               const float* __restrict__ sq,
                               int* __restrict__ nn) {
  __shared__ float dist[ROWS * LDSP];

  const int lane = threadIdx.x;
  const int b    = blockIdx.x / (NPTS / ROWS);
  const int rb   = blockIdx.x % (NPTS / ROWS);
  const int row0 = rb * ROWS;

  const float* xb  = x  + (size_t)b * NPTS * DIM;
  const float* sqb = sq + (size_t)b * NPTS;

  const int lmod  = lane & 15;        // M (A) / N (B) index within tile
  const int khalf = (lane >> 4) * 2;  // K sub-offset: lanes 16-31 carry K+2,K+3

  // --- A fragments for 16 K-steps, rows [row0..row0+15] and [row0+16..row0+31]
  v2f a0[16], a1[16];
  {
    const float* r0 = xb + (size_t)(row0 + lmod) * DIM + khalf;
    const float* r1 = xb + (size_t)(row0 + 16 + lmod) * DIM + khalf;
#pragma unroll
    for (int k = 0; k < 16; ++k) {
      a0[k] = *(const v2f*)(r0 + 4 * k);
      a1[k] = *(const v2f*)(r1 + 4 * k);
    }
  }

  // --- per-lane sorted top-20 (ascending key; stable => lower index wins ties)
  float bd[KNN];
  int   bi[KNN];
#pragma unroll
  for (int j = 0; j < KNN; ++j) { bd[j] = __builtin_inff(); bi[j] = 0x7fffffff; }

  // prefetch B fragments for tile t (clamped) into a register buffer
  auto prefetch = [&](v2f (&buf)[16], float& sqc, int t) {
    int tt = (t < NTILES) ? t : (NTILES - 1);
    const float* p = xb + (size_t)(tt * CTILE + lmod) * DIM + khalf;
#pragma unroll
    for (int k = 0; k < 16; ++k) buf[k] = *(const v2f*)(p + 4 * k);
    sqc = sqb[tt * CTILE + lmod];
  };

  // WMMA + LDS transpose + stable top-20 insertion for one 32x16 tile
  auto process = [&](const v2f (&buf)[16], float sqcol, int t) {
    const int col0 = t * CTILE;
    v8f acc0 = {};
    v8f acc1 = {};
#pragma unroll
    for (int k = 0; k < 16; ++k) {
      acc0 = __builtin_amdgcn_wmma_f32_16x16x4_f32(false, a0[k], false, buf[k],
                                                   (short)0, acc0, false, false);
      acc1 = __builtin_amdgcn_wmma_f32_16x16x4_f32(false, a1[k], false, buf[k],
                                                   (short)0, acc1, false, false);
    }
    // C layout: lanes 0-15 vgpr r -> (M=r, N=lane); lanes 16-31 -> (M=8+r, N=lane-16)
    const int rbase = (lane >> 4) * 8;
#pragma unroll
    for (int r = 0; r < 8; ++r) {
      dist[(rbase + r) * LDSP + lmod]      = sqcol - 2.0f * acc0[r];
      dist[(16 + rbase + r) * LDSP + lmod] = sqcol - 2.0f * acc1[r];
    }
    __syncthreads();

    // lane owns row (row0 + lane): scan this tile's 16 candidates
#pragma unroll 1
    for (int c = 0; c < CTILE; ++c) {
      float cd = dist[lane * LDSP + c];
      if (cd < bd[KNN - 1]) {           // strict: equal key keeps earlier index
        int ci = col0 + c;
#pragma unroll
        for (int j = 0; j < KNN; ++j) { // stable bubble-insert, all in registers
          bool sm  = cd < bd[j];
          float td = bd[j];
          int   ti = bi[j];
          bd[j] = sm ? cd : bd[j];
          bi[j] = sm ? ci : bi[j];
          cd = sm ? td : cd;
          ci = sm ? ti : ci;
        }
      }
    }
    __syncthreads();
  };

  // software-pipelined double buffer: loads for tile t+1 issue before the
  // WMMA + top-k work of tile t, hiding L2 latency with no copy-wait point
  v2f bufA[16], bufB[16];
  float sqA, sqB;
  prefetch(bufA, sqA, 0);
#pragma unroll 1
  for (int t = 0; t < NTILES; t += 2) {
    prefetch(bufB, sqB, t + 1);
    process(bufA, sqA, t);
    prefetch(bufA, sqA, t + 2);
    process(bufB, sqB, t + 1);
  }

  int* out = nn + ((size_t)b * NPTS + row0 + lane) * KNN;
#pragma unroll
  for (int j = 0; j < KNN; ++j) out[j] = bi[j];
}

// ---------------- kernel 2: gather + concat(central, neighbor-central) ----------------
// One float4 per thread; non-temporal b128 stores keep x/nn L2-resident.
__global__ void EdgeGather_kernel(const float* __restrict__ x,
                                  const int* __restrict__ nn,
                                  float* __restrict__ out) {
  unsigned gid = blockIdx.x * blockDim.x + threadIdx.x;
  const unsigned total4 = (unsigned)BATCH * NPTS * KNN * (2 * DIM / 4);
  if (gid >= total4) return;

  unsigned c4 = gid & (2 * DIM / 4 - 1); // 0..31: float4 slot within 128-float row
  unsigned nk = gid >> 5;                // (b*N + n)*KNN + k
  unsigned bn = nk / KNN;
  unsigned n  = bn & (NPTS - 1);
  unsigned b  = bn >> 12;

  const v4f* x4 = (const v4f*)(x + (size_t)b * NPTS * DIM);
  v4f v;
  if (c4 < DIM / 4) {
    v = x4[n * (DIM / 4) + c4];
  } else {
    unsigned j = (unsigned)nn[nk];
    unsigned d = c4 - DIM / 4;
    v = x4[j * (DIM / 4) + d] - x4[n * (DIM / 4) + d];
  }
  __builtin_nontemporal_store(v, (v4f*)out + gid);
}

extern "C" void kernel_launch(void* const* d_in, const int* in_sizes, int n_in,
                              void* d_out, int out_size, void* d_ws, size_t ws_size,
                              hipStream_t stream) {
  const float* x = (const float*)d_in[0];
  float* sq = (float*)d_ws;                                               // 128 KB
  int*   nn = (int*)((char*)d_ws + (size_t)BATCH * NPTS * sizeof(float)); // 2.5 MB
  float* out = (float*)d_out;

  EdgeSq_kernel<<<(BATCH * NPTS + 255) / 256, 256, 0, stream>>>(x, sq);
  EdgeKnn_kernel<<<BATCH * (NPTS / ROWS), 32, 0, stream>>>(x, sq, nn);

  unsigned total4 = (unsigned)BATCH * NPTS * KNN * (2 * DIM / 4);
  EdgeGather_kernel<<<(total4 + 255) / 256, 256, 0, stream>>>(x, nn, out);
}